// NeocorticalModule_24043226923366
// MI455X (gfx1250) — compile-verified
//
#include <hip/hip_runtime.h>
#include <hip/hip_bf16.h>
#include <math.h>

#define DIMK   1024
#define HID    64      // 2*SCHEMA_DIM
#define SD     32      // SCHEMA_DIM
#define NS     64      // NUM_SCHEMAS
#define NTOT   131072
#define LRATE  0.01f
#define EPSI   1e-8f

#define ROWS    128    // rows per workgroup
#define THREADS 128    // 4 wave32 waves, 32 rows per wave
#define KCHUNK  32     // K-chunk staged in LDS

typedef float v2f __attribute__((ext_vector_type(2)));
typedef float v8f __attribute__((ext_vector_type(8)));

__global__ void zero_ws_kernel(float* __restrict__ p, int n) {
    int i = blockIdx.x * blockDim.x + threadIdx.x;
    for (; i < n; i += gridDim.x * blockDim.x) p[i] = 0.0f;
}

__launch_bounds__(THREADS, 2)
__global__ void fused_encode_assign_kernel(
    const float* __restrict__ X,       // (N, 1024)
    const float* __restrict__ W1,      // (1024, 64)
    const float* __restrict__ b1,      // (64,)
    const float* __restrict__ W2,      // (64, 32)
    const float* __restrict__ b2,      // (32,)
    const float* __restrict__ schemas, // (64, 32)
    float* __restrict__ g_sums,        // (64*32,) accumulators
    float* __restrict__ g_counts)      // (64,)
{
    __shared__ float sXt[ROWS][KCHUNK];      // 16 KB  X tile
    __shared__ v2f   sW1p[KCHUNK / 2][HID];  //  8 KB  W1 chunk, K-pair interleaved
    __shared__ v2f   sW2p[HID / 2][SD];      //  8 KB  full W2, K-pair interleaved
    __shared__ float sSchN[NS][SD];          //  8 KB  normalized schemas
    __shared__ float sH[ROWS][HID];          // 32 KB  hidden strip
    __shared__ float sE[ROWS][SD];           // 16 KB  encoded strip
    __shared__ float sAcc[NS * SD];          //  8 KB  block-local scatter sums
    __shared__ float sCnt[NS];

    const int tid  = threadIdx.x;
    const int wave = tid >> 5;
    const int lane = tid & 31;
    const int l15  = lane & 15;
    const int hi   = lane >> 4;              // 0: lanes 0-15, 1: lanes 16-31
    const int row0 = blockIdx.x * ROWS;

    // ---- stage W2 in K-pair layout: sW2p[k/2][n] = {W2[k][n], W2[k+1][n]} ----
    #pragma unroll
    for (int v = 0; v < 4; ++v) {
        int idx = v * 512 + tid * 4;         // 2048 floats, coalesced float4 reads
        int r = idx >> 5, c = idx & 31;
        float4 w = *(const float4*)&W2[(size_t)r * SD + c];
        float* dst = (float*)&sW2p[r >> 1][c];
        int sel = r & 1;
        dst[sel] = w.x; dst[2 + sel] = w.y; dst[4 + sel] = w.z; dst[6 + sel] = w.w;
    }
    // ---- normalized schemas (argmax of cosine == argmax of dot vs normalized schemas) ----
    if (tid < NS) {
        float nsq = 0.0f;
        #pragma unroll
        for (int d = 0; d < SD; ++d) { float v = schemas[tid * SD + d]; nsq += v * v; }
        float inv = 1.0f / fmaxf(sqrtf(nsq), EPSI);
        #pragma unroll
        for (int d = 0; d < SD; ++d) sSchN[tid][d] = schemas[tid * SD + d] * inv;
    }
    // ---- zero block-local scatter accumulators ----
    #pragma unroll
    for (int v = 0; v < 16; ++v) sAcc[tid * 16 + v] = 0.0f;
    if (tid < NS) sCnt[tid] = 0.0f;

    // ============ GEMM1: h = relu(X @ W1 + b1) via V_WMMA_F32_16X16X4_F32 ============
    // Each wave owns 2 row-tiles (32 rows) x 4 col-tiles: B frags reused across 2 A frags.
    v8f acc[2][4];
    #pragma unroll
    for (int t = 0; t < 2; ++t)
        #pragma unroll
        for (int n = 0; n < 4; ++n)
            #pragma unroll
            for (int j = 0; j < 8; ++j) acc[t][n][j] = 0.0f;

    for (int k0 = 0; k0 < DIMK; k0 += KCHUNK) {
        __syncthreads();   // previous chunk's consumers done
        // stage X tile: 128x32 floats; consecutive lanes -> consecutive 16B (coalesced)
        #pragma unroll
        for (int v = 0; v < 8; ++v) {
            int idx = v * 512 + tid * 4;
            int r = idx >> 5, c = idx & 31;
            *(float4*)&sXt[r][c] = *(const float4*)&X[(size_t)(row0 + r) * DIMK + k0 + c];
        }
        // stage W1 chunk in K-pair layout: sW1p[k/2][n] = {W1[k][n], W1[k+1][n]}
        #pragma unroll
        for (int v = 0; v < 4; ++v) {
            int idx = v * 512 + tid * 4;     // 2048 floats, coalesced float4 reads
            int r = idx >> 6, c = idx & 63;
            float4 w = *(const float4*)&W1[(size_t)(k0 + r) * HID + c];
            float* dst = (float*)&sW1p[r >> 1][c];
            int sel = r & 1;
            dst[sel] = w.x; dst[2 + sel] = w.y; dst[4 + sel] = w.z; dst[6 + sel] = w.w;
        }
        // prefetch next chunk: one line per row covers the whole next X tile
        if (k0 + KCHUNK < DIMK)
            __builtin_prefetch(&X[(size_t)(row0 + tid) * DIMK + k0 + KCHUNK], 0, 0);
        __syncthreads();   // tiles visible

        #pragma unroll
        for (int kk = 0; kk < KCHUNK; kk += 4) {
            const int kof = kk + hi * 2;     // A/B K index for this half-wave
            // A 16x4 frags: lanes 0-15 K={0,1}, lanes 16-31 K={2,3} (contiguous b64)
            v2f a0 = *(const v2f*)&sXt[wave * 32 + l15][kof];
            v2f a1 = *(const v2f*)&sXt[wave * 32 + 16 + l15][kof];
            #pragma unroll
            for (int n = 0; n < 4; ++n) {
                // B 4x16 frag: single b64, pair-interleaved so no repack movs
                v2f b = sW1p[(kk >> 1) + hi][n * 16 + l15];
                acc[0][n] = __builtin_amdgcn_wmma_f32_16x16x4_f32(
                    false, a0, false, b, (short)0, acc[0][n], false, false);
                acc[1][n] = __builtin_amdgcn_wmma_f32_16x16x4_f32(
                    false, a1, false, b, (short)0, acc[1][n], false, false);
            }
        }
    }

    // bias + ReLU, store 32x64 strip (C/D layout: VGPR j -> rows j and j+8)
    #pragma unroll
    for (int t = 0; t < 2; ++t)
        #pragma unroll
        for (int n = 0; n < 4; ++n) {
            float bias = b1[n * 16 + l15];
            #pragma unroll
            for (int j = 0; j < 8; ++j) {
                float v = fmaxf(acc[t][n][j] + bias, 0.0f);
                sH[wave * 32 + t * 16 + j + hi * 8][n * 16 + l15] = v;
            }
        }
    __syncthreads();

    // ============ GEMM2: encoded = h @ W2 + b2 (M=32/wave, K=64, N=32) ============
    v8f acc2[2][2];
    #pragma unroll
    for (int t = 0; t < 2; ++t)
        #pragma unroll
        for (int n = 0; n < 2; ++n)
            #pragma unroll
            for (int j = 0; j < 8; ++j) acc2[t][n][j] = 0.0f;

    #pragma unroll
    for (int kk = 0; kk < HID; kk += 4) {
        const int kof = kk + hi * 2;
        v2f a0 = *(const v2f*)&sH[wave * 32 + l15][kof];
        v2f a1 = *(const v2f*)&sH[wave * 32 + 16 + l15][kof];
        #pragma unroll
        for (int n = 0; n < 2; ++n) {
            v2f b = sW2p[(kk >> 1) + hi][n * 16 + l15];
            acc2[0][n] = __builtin_amdgcn_wmma_f32_16x16x4_f32(
                false, a0, false, b, (short)0, acc2[0][n], false, false);
            acc2[1][n] = __builtin_amdgcn_wmma_f32_16x16x4_f32(
                false, a1, false, b, (short)0, acc2[1][n], false, false);
        }
    }
    #pragma unroll
    for (int t = 0; t < 2; ++t)
        #pragma unroll
        for (int n = 0; n < 2; ++n) {
            float bias = b2[n * 16 + l15];
            #pragma unroll
            for (int j = 0; j < 8; ++j)
                sE[wave * 32 + t * 16 + j + hi * 8][n * 16 + l15] = acc2[t][n][j] + bias;
        }
    __syncthreads();

    // ============ per-row argmax assignment + LDS scatter-accumulate ============
    {
        const int r = tid;                  // 128 threads, 128 rows: all lanes active
        float e[SD];
        #pragma unroll
        for (int d = 0; d < SD; ++d) e[d] = sE[r][d];
        float best = -3.402823466e+38f;
        int bi = 0;
        for (int s = 0; s < NS; ++s) {
            float dot = 0.0f;
            #pragma unroll
            for (int d = 0; d < SD; ++d) dot += e[d] * sSchN[s][d];
            if (dot > best) { best = dot; bi = s; }   // first-max tie break (matches argmax)
        }
        #pragma unroll
        for (int d = 0; d < SD; ++d) atomicAdd(&sAcc[bi * SD + d], e[d]);
        atomicAdd(&sCnt[bi], 1.0f);
    }
    __syncthreads();

    // ============ one global flush per block ============
    #pragma unroll
    for (int v = 0; v < 16; ++v) {
        int i = tid * 16 + v;
        atomicAdd(&g_sums[i], sAcc[i]);
    }
    if (tid < NS) atomicAdd(&g_counts[tid], sCnt[tid]);
}

__global__ void finalize_kernel(
    const float* __restrict__ g_sums,
    const float* __restrict__ g_counts,
    const float* __restrict__ schemas,
    const float* __restrict__ usage,
    float* __restrict__ out)   // [0..2047]=new_schemas [2048..2111]=new_usage [2112]=num_updated [2113]=mean_norm
{
    __shared__ float sN[NS];
    __shared__ int   sU[NS];
    const int s = threadIdx.x;   // 64 threads
    float cnt = g_counts[s];
    int   ne  = cnt > 0.0f ? 1 : 0;
    float inv = 1.0f / fmaxf(cnt, 1.0f);
    float nsq = 0.0f;
    #pragma unroll
    for (int d = 0; d < SD; ++d) {
        float sc     = schemas[s * SD + d];
        float target = g_sums[s * SD + d] * inv;
        float delta  = LRATE * (target - sc) * (float)ne;
        out[s * SD + d] = sc + delta;
        nsq += delta * delta;
    }
    out[NS * SD + s] = usage[s] + cnt;
    sN[s] = sqrtf(nsq);
    sU[s] = ne;
    __syncthreads();
    if (s == 0) {
        float sum = 0.0f; int nu = 0;
        for (int i = 0; i < NS; ++i) { sum += sN[i]; nu += sU[i]; }
        out[NS * SD + NS]     = (float)nu;
        out[NS * SD + NS + 1] = sum / (float)(nu > 1 ? nu : 1);
    }
}

extern "C" void kernel_launch(void* const* d_in, const int* in_sizes, int n_in,
                              void* d_out, int out_size, void* d_ws, size_t ws_size,
                              hipStream_t stream) {
    const float* X       = (const float*)d_in[0];  // episodic_traces (131072,1024)
    const float* W1      = (const float*)d_in[1];  // (1024,64)
    const float* b1      = (const float*)d_in[2];  // (64,)
    const float* W2      = (const float*)d_in[3];  // (64,32)
    const float* b2      = (const float*)d_in[4];  // (32,)
    const float* schemas = (const float*)d_in[5];  // (64,32)
    const float* usage   = (const float*)d_in[6];  // (64,)
    float* out = (float*)d_out;

    float* g_sums   = (float*)d_ws;          // 64*32
    float* g_counts = g_sums + NS * SD;      // 64

    zero_ws_kernel<<<1, 256, 0, stream>>>(g_sums, NS * SD + NS);

    fused_encode_assign_kernel<<<NTOT / ROWS, THREADS, 0, stream>>>(
        X, W1, b1, W2, b2, schemas, g_sums, g_counts);

    finalize_kernel<<<1, NS, 0, stream>>>(g_sums, g_counts, schemas, usage, out);
}